// PCENLayer_24318104830260
// MI455X (gfx1250) — compile-verified
//
#include <hip/hip_runtime.h>

// PCEN for x[32,1,128,8000] f32.
// One workgroup (256 threads = 8 waves) per (B,F) row.
// - Row staged into LDS via global_load_async_to_lds_b128 (ASYNCcnt path)
// - Local scans (16-step chunks) via chained v_wmma_f32_16x16x4_f32
// - 2-level carry scan across 512 chunks
// - Pointwise PCEN vectorized: ds_load_b128 + global_store_b128 (float4/thread),
//   loop kept rolled to avoid I-cache bloat (transcendental-heavy body)

#define T_LEN   8000
#define T_PAD   8192          // 512 chunks of 16
#define NCHUNK  512
#define NBANDS  128
#define PCEN_EPS 1e-6f

typedef float v2f __attribute__((ext_vector_type(2)));
typedef float v8f __attribute__((ext_vector_type(8)));

static __device__ __forceinline__ uint32_t lds_off(const void* p) {
    // generic LDS pointer: low 32 bits are the LDS byte offset
    return (uint32_t)(uintptr_t)p;
}

__global__ __launch_bounds__(256)
void pcen_wmma_kernel(const float* __restrict__ x,
                      const float* __restrict__ log_s,
                      const float* __restrict__ log_alpha,
                      const float* __restrict__ log_delta,
                      const float* __restrict__ log_r,
                      float* __restrict__ out)
{
    __shared__ __align__(16) float sx[T_PAD];   // input row (padded with zeros)
    __shared__ __align__(16) float sf[T_PAD];   // local-scan results
    __shared__ float chunkLast[NCHUNK];         // local[15] of each 16-chunk
    __shared__ float carries[NCHUNK];           // carry INTO each chunk
    __shared__ float segOff[32];
    __shared__ float sApow[16];                 // a^0 .. a^15

    const int row  = blockIdx.x;                // 0..4095
    const int band = row & (NBANDS - 1);        // F index
    const int tid  = threadIdx.x;
    const float* __restrict__ xr = x + (size_t)row * T_LEN;
    float* __restrict__ yr = out + (size_t)row * T_LEN;

    // per-band constants (scalar broadcast loads)
    const float s     = __expf(log_s[band]);
    const float a     = 1.0f - s;
    const float alpha = __expf(log_alpha[band]);
    const float r_    = __expf(log_r[band]);
    const float delta = __expf(log_delta[band]);
    const float dr    = __expf(r_ * log_delta[band]);   // delta^r

    // ---- async-stage row into LDS (16B per lane per round) ----
    {
        const uint64_t gbase  = (uint64_t)(uintptr_t)xr;   // 16B aligned (row*32000)
        const uint32_t sxbase = lds_off(&sx[0]);
        #pragma unroll
        for (int m = 0; m < 8; ++m) {
            if (m < 7 || tid < 208) {           // round 7: floats 7168..7999 only
                const uint32_t voff = (uint32_t)(m * 4096 + tid * 16);
                const uint32_t ldsv = sxbase + voff;
                asm volatile("global_load_async_to_lds_b128 %0, %1, %2"
                             :: "v"(ldsv), "v"(voff), "s"(gbase) : "memory");
            }
        }
    }
    // power table in LDS (dynamic-indexable without select chains)
    if (tid == 0) {
        float p = 1.0f;
        #pragma unroll
        for (int d = 0; d < 16; ++d) { sApow[d] = p; p *= a; }
    }
    if (tid < T_PAD - T_LEN)                    // zero pad sx[8000..8191]
        sx[T_LEN + tid] = 0.0f;
    asm volatile("s_wait_asynccnt 0x0" ::: "memory");
    __syncthreads();

    // ---- per-lane A-matrix fragments: M[t][j] = s * a^(t-j) (lower-tri) ----
    const int lane = tid & 31;
    const int hi   = lane >> 4;      // 0: lanes 0-15, 1: lanes 16-31
    const int mrow = lane & 15;      // A row / B,D column index

    v2f Afrag[4];
    #pragma unroll
    for (int kk = 0; kk < 4; ++kk) {
        const int c0 = 4 * kk + 2 * hi;           // K-columns (c0, c0+1)
        const int d0 = mrow - c0;
        const int d1 = d0 - 1;
        const float v0 = s * sApow[d0 < 0 ? 0 : d0];
        const float v1 = s * sApow[d1 < 0 ? 0 : d1];
        Afrag[kk].x = (d0 >= 0) ? v0 : 0.0f;
        Afrag[kk].y = (d1 >= 0) ? v1 : 0.0f;
    }

    // ---- local scans: 32 tiles of 256 elems; 4 tiles per wave ----
    const int wave = tid >> 5;
    #pragma unroll
    for (int j = 0; j < 4; ++j) {
        const int tile = wave * 4 + j;
        const int base = tile * 256;
        v8f acc = {};
        #pragma unroll
        for (int kk = 0; kk < 4; ++kk) {
            const int j0 = 4 * kk + 2 * hi;       // B K-rows (j0, j0+1)
            v2f b;
            b.x = sx[base + mrow * 16 + j0];      // X[j0][n],   n = mrow
            b.y = sx[base + mrow * 16 + j0 + 1];  // X[j0+1][n]
            acc = __builtin_amdgcn_wmma_f32_16x16x4_f32(
                    false, Afrag[kk], false, b, (short)0, acc, false, false);
        }
        // D layout: VGPR v, this lane -> t = v + 8*hi, column n = mrow
        #pragma unroll
        for (int v = 0; v < 8; ++v)
            sf[base + mrow * 16 + (v + 8 * hi)] = acc[v];
        if (hi)
            chunkLast[tile * 16 + mrow] = acc[7]; // t == 15
    }
    __syncthreads();

    // ---- carry scan across 512 chunks: F_k = a^16 * F_{k-1} + local[15] ----
    float a16 = a * a;                            // a^2
    a16 = a16 * a16;                              // a^4
    a16 = a16 * a16;                              // a^8
    a16 = a16 * a16;                              // a^16
    float a256 = a16 * a16;                       // a^32
    a256 = a256 * a256;                           // a^64
    a256 = a256 * a256;                           // a^128
    a256 = a256 * a256;                           // a^256

    if (tid < 32) {                               // level 1: 16 chunks per lane
        float o = 0.0f;
        #pragma unroll
        for (int m = 0; m < 16; ++m)
            o = a16 * o + chunkLast[tid * 16 + m];
        segOff[tid] = o;
    }
    __syncthreads();
    if (tid == 0) {                               // level 2: 32-step serial
        float F = sx[0];                          // F_{-1} = x[0] => f[0]=x[0]
        for (int l = 0; l < 32; ++l) {
            const float nf = a256 * F + segOff[l];
            segOff[l] = F;                        // carry into segment l
            F = nf;
        }
    }
    __syncthreads();
    if (tid < 32) {                               // level 3: expand carries
        float F = segOff[tid];
        #pragma unroll
        for (int m = 0; m < 16; ++m) {
            carries[tid * 16 + m] = F;
            F = a16 * F + chunkLast[tid * 16 + m];
        }
    }
    __syncthreads();

    // ---- pointwise PCEN + vectorized store (float4 per thread per round) ----
    // element i: filtered = sf[i] + a^((i&15)+1) * carries[i>>4]
    // per-thread t0 = (4*tid)&15 is invariant across rounds (stride 1024 = 0 mod 16);
    // a float4 never straddles a 16-chunk boundary (4*tid % 16 <= 12).
    const float at1 = sApow[(4 * tid) & 15] * a;  // a^(t0+1)
    const float k0 = at1, k1 = at1 * a, k2 = k1 * a, k3 = k2 * a;

    for (int m = 0; m < 8; ++m) {                 // rolled: keep I-cache small
        const int i = m * 1024 + tid * 4;
        if (i < T_LEN) {                          // round 7: tid < 208
            const float4 fv = *(const float4*)&sf[i];
            const float4 xv = *(const float4*)&sx[i];
            const float  cr = carries[i >> 4];    // same chunk for all 4 elems

            const float f0 = fv.x + k0 * cr;
            const float f1 = fv.y + k1 * cr;
            const float f2 = fv.z + k2 * cr;
            const float f3 = fv.w + k3 * cr;

            float4 yv;
            yv.x = __expf(r_ * __logf(xv.x * __expf(-alpha * __logf(PCEN_EPS + f0)) + delta)) - dr;
            yv.y = __expf(r_ * __logf(xv.y * __expf(-alpha * __logf(PCEN_EPS + f1)) + delta)) - dr;
            yv.z = __expf(r_ * __logf(xv.z * __expf(-alpha * __logf(PCEN_EPS + f2)) + delta)) - dr;
            yv.w = __expf(r_ * __logf(xv.w * __expf(-alpha * __logf(PCEN_EPS + f3)) + delta)) - dr;

            *(float4*)&yr[i] = yv;                // global_store_b128
        }
    }
}

extern "C" void kernel_launch(void* const* d_in, const int* in_sizes, int n_in,
                              void* d_out, int out_size, void* d_ws, size_t ws_size,
                              hipStream_t stream) {
    const float* x         = (const float*)d_in[0];
    const float* log_s     = (const float*)d_in[1];
    const float* log_alpha = (const float*)d_in[2];
    const float* log_delta = (const float*)d_in[3];
    const float* log_r     = (const float*)d_in[4];
    float* out = (float*)d_out;

    const int rows = in_sizes[0] / T_LEN;         // 4096
    hipLaunchKernelGGL(pcen_wmma_kernel, dim3(rows), dim3(256), 0, stream,
                       x, log_s, log_alpha, log_delta, log_r, out);
}